// LeViT_47888885350866
// MI455X (gfx1250) — compile-verified
//
#include <hip/hip_runtime.h>
#include <math.h>

// ---------------------------------------------------------------------------
// LeViT forward for MI455X (gfx1250).  All GEMMs go through the CDNA5 f32
// matrix pipe: V_WMMA_F32_16X16X4_F32 (wave32, 16x16 tile per wave-issue,
// 8-VGPR f32 accumulator, fragment layouts per ISA 7.12.2).  Model is
// memory-bound (~25 GFLOP vs ~0.3 GB traffic at 23.3 TB/s) so we keep f32.
// Round-2 refinements: 64x64 block tile, 16x32 per wave (2 accumulators
// sharing one A fragment -> 8 WMMA per staged chunk), interior fast-path
// staging without per-element guards, LDS padding (17) to kill the 4-way
// bank conflict on A-fragment reads.
// ---------------------------------------------------------------------------

typedef float v2f __attribute__((ext_vector_type(2)));
typedef float v8f __attribute__((ext_vector_type(8)));

#define TBM 64
#define TBN 64
#define TBK 16
#define APAD (TBK + 1)   // 17, coprime with 64 banks

__device__ __forceinline__ float hswish_f(float v) {
    float t = fminf(fmaxf(v + 3.0f, 0.0f), 6.0f);
    return v * t * (1.0f / 6.0f);
}

// Generalized batched GEMM:  C[g] = alpha * A[g] @ op(B[g]) (+ biasM / biasV)
// g = bb*heads + hh ; base += bb*s?b + hh*s?h.  transB: B stored [N,K] row-major.
__launch_bounds__(256)
__global__ void gemm_f32_wmma(const float* __restrict__ A,
                              const float* __restrict__ B,
                              float* __restrict__ C,
                              int M, int N, int K,
                              int lda, int ldb, int ldc, int heads,
                              long sAb, long sAh, long sBb, long sBh,
                              long sCb, long sCh,
                              int transB, float alpha,
                              const float* __restrict__ biasM, int biasLd,
                              const float* __restrict__ biasV)
{
    __shared__ float sA[TBM * APAD];
    __shared__ float sB[TBK * TBN];

    int g  = blockIdx.z;
    int bb = g / heads, hh = g - bb * heads;
    A += (long)bb * sAb + (long)hh * sAh;
    B += (long)bb * sBb + (long)hh * sBh;
    C += (long)bb * sCb + (long)hh * sCh;

    int tile_m = blockIdx.y * TBM;
    int tile_n = blockIdx.x * TBN;
    int tid  = threadIdx.x;
    int wave = tid >> 5;
    int lane = tid & 31;
    int wm = wave & 3;             // 0..3  (M sub-tile, 16 rows each)
    int wn = wave >> 2;            // 0..1  (N sub-tile, 32 cols each)
    int m0 = wm * 16;
    int n0 = wn * 32;
    int half = lane >> 4;          // 0: lanes 0-15, 1: lanes 16-31
    int l16  = lane & 15;

    v8f acc0 = {};
    v8f acc1 = {};

    for (int k0 = 0; k0 < K; k0 += TBK) {
        // ---- Stage A tile (64 x 16, padded rows)
        if (tile_m + TBM <= M && k0 + TBK <= K) {
            for (int i = tid; i < TBM * TBK; i += 256) {
                int r = i >> 4, c = i & 15;
                sA[r * APAD + c] = A[(long)(tile_m + r) * lda + (k0 + c)];
            }
        } else {
            for (int i = tid; i < TBM * TBK; i += 256) {
                int r = i >> 4, c = i & 15;
                int gm = tile_m + r, gk = k0 + c;
                sA[r * APAD + c] = (gm < M && gk < K) ? A[(long)gm * lda + gk] : 0.0f;
            }
        }
        // ---- Stage B tile (16 x 64)
        if (tile_n + TBN <= N && k0 + TBK <= K) {
            if (transB) {
                for (int i = tid; i < TBN * TBK; i += 256) {
                    int r = i >> 4, c = i & 15;   // r: n, c: k
                    sB[c * TBN + r] = B[(long)(tile_n + r) * ldb + (k0 + c)];
                }
            } else {
                for (int i = tid; i < TBK * TBN; i += 256) {
                    int r = i >> 6, c = i & 63;   // r: k, c: n
                    sB[r * TBN + c] = B[(long)(k0 + r) * ldb + (tile_n + c)];
                }
            }
        } else {
            for (int i = tid; i < TBK * TBN; i += 256) {
                int r = i >> 6, c = i & 63;       // r: k, c: n
                int gk = k0 + r, gn = tile_n + c;
                float v = 0.0f;
                if (gk < K && gn < N)
                    v = transB ? B[(long)gn * ldb + gk] : B[(long)gk * ldb + gn];
                sB[r * TBN + c] = v;
            }
        }
        __syncthreads();

        // 8 x V_WMMA_F32_16X16X4_F32 per staged chunk (A fragment reused 2x).
        // A-frag (16x4): lanes 0-15 hold K=kk..kk+1, lanes 16-31 K=kk+2..kk+3.
        // B-frag (4x16): VGPR0 = rows {kk, kk+2}, VGPR1 = rows {kk+1, kk+3}.
        #pragma unroll
        for (int kk = 0; kk < TBK; kk += 4) {
            v2f af, bf0, bf1;
            af.x  = sA[(m0 + l16) * APAD + kk + 2 * half + 0];
            af.y  = sA[(m0 + l16) * APAD + kk + 2 * half + 1];
            bf0.x = sB[(kk + 2 * half + 0) * TBN + n0 + l16];
            bf0.y = sB[(kk + 2 * half + 1) * TBN + n0 + l16];
            bf1.x = sB[(kk + 2 * half + 0) * TBN + n0 + 16 + l16];
            bf1.y = sB[(kk + 2 * half + 1) * TBN + n0 + 16 + l16];
            acc0 = __builtin_amdgcn_wmma_f32_16x16x4_f32(
                       false, af, false, bf0, (short)0, acc0, false, false);
            acc1 = __builtin_amdgcn_wmma_f32_16x16x4_f32(
                       false, af, false, bf1, (short)0, acc1, false, false);
        }
        __syncthreads();
    }

    // C/D layout: VGPR v -> M = m0+v (+8 for upper half-lanes), N = n0+l16.
    int gn0 = tile_n + n0 + l16;
    int gn1 = gn0 + 16;
    #pragma unroll
    for (int v = 0; v < 8; v++) {
        int gm = tile_m + m0 + v + 8 * half;
        if (gm < M) {
            if (gn0 < N) {
                float r = acc0[v] * alpha;
                if (biasM) r += biasM[(long)gm * biasLd + gn0];
                if (biasV) r += biasV[gn0];
                C[(long)gm * ldc + gn0] = r;
            }
            if (gn1 < N) {
                float r = acc1[v] * alpha;
                if (biasM) r += biasM[(long)gm * biasLd + gn1];
                if (biasV) r += biasV[gn1];
                C[(long)gm * ldc + gn1] = r;
            }
        }
    }
}

// Direct 3x3 stride-2 NHWC conv, SAME padding (pad_before = 0 for even H here).
__global__ void conv3x3s2(const float* __restrict__ X, const float* __restrict__ Wt,
                          const float* __restrict__ Bs, float* __restrict__ Y,
                          int Bn, int H, int W, int Cin, int OH, int OW, int Cout)
{
    long i = (long)blockIdx.x * blockDim.x + threadIdx.x;
    long total = (long)Bn * OH * OW * Cout;
    if (i >= total) return;
    int oc = (int)(i % Cout);
    long t = i / Cout;
    int ox = (int)(t % OW); t /= OW;
    int oy = (int)(t % OH); t /= OH;
    int b  = (int)t;
    float acc = Bs[oc];
    int iy0 = oy * 2, ix0 = ox * 2;
    for (int ky = 0; ky < 3; ky++) {
        int iy = iy0 + ky;
        if (iy >= H) break;
        for (int kx = 0; kx < 3; kx++) {
            int ix = ix0 + kx;
            if (ix >= W) continue;
            const float* xp = X + (((long)b * H + iy) * W + ix) * Cin;
            const float* wp = Wt + (long)((ky * 3 + kx) * Cin) * Cout + oc;
            for (int c = 0; c < Cin; c++)
                acc += xp[c] * wp[(long)c * Cout];
        }
    }
    Y[i] = acc;
}

// Per-column mean / rstd over M rows of X[M,C] (BN over all leading axes).
__global__ void colstats(const float* __restrict__ X, int M, int C,
                         float* __restrict__ mean, float* __restrict__ rstd)
{
    int c = blockIdx.x;
    float s = 0.0f, s2 = 0.0f;
    for (int m = threadIdx.x; m < M; m += blockDim.x) {
        float v = X[(long)m * C + c];
        s += v; s2 += v * v;
    }
    __shared__ float sh0[256];
    __shared__ float sh1[256];
    sh0[threadIdx.x] = s; sh1[threadIdx.x] = s2;
    __syncthreads();
    for (int off = 128; off > 0; off >>= 1) {
        if ((int)threadIdx.x < off) {
            sh0[threadIdx.x] += sh0[threadIdx.x + off];
            sh1[threadIdx.x] += sh1[threadIdx.x + off];
        }
        __syncthreads();
    }
    if (threadIdx.x == 0) {
        float mu  = sh0[0] / (float)M;
        float var = sh1[0] / (float)M - mu * mu;
        mean[c] = mu;
        rstd[c] = rsqrtf(var + 1e-5f);
    }
}

// In-place BN apply (+optional bias, +optional hard-swish).
__global__ void bn_act(float* __restrict__ X, long n, int C,
                       const float* __restrict__ mean, const float* __restrict__ rstd,
                       const float* __restrict__ scale, const float* __restrict__ bias,
                       int act)
{
    long i = (long)blockIdx.x * blockDim.x + threadIdx.x;
    if (i >= n) return;
    int c = (int)(i % C);
    float v = (X[i] - mean[c]) * rstd[c] * scale[c];
    if (bias) v += bias[c];
    if (act)  v = hswish_f(v);
    X[i] = v;
}

__global__ void hswish_k(float* __restrict__ X, long n)
{
    long i = (long)blockIdx.x * blockDim.x + threadIdx.x;
    if (i < n) X[i] = hswish_f(X[i]);
}

__global__ void add_k(float* __restrict__ X, const float* __restrict__ T, long n)
{
    long i = (long)blockIdx.x * blockDim.x + threadIdx.x;
    if (i < n) X[i] += T[i];
}

// Row softmax in place, one workgroup per row.
__global__ void softmax_rows(float* __restrict__ X, int L)
{
    float* row = X + (long)blockIdx.x * L;
    int tid = threadIdx.x;
    __shared__ float sh[256];
    float mx = -3.4e38f;
    for (int j = tid; j < L; j += blockDim.x) mx = fmaxf(mx, row[j]);
    sh[tid] = mx;
    __syncthreads();
    for (int off = 128; off > 0; off >>= 1) {
        if (tid < off) sh[tid] = fmaxf(sh[tid], sh[tid + off]);
        __syncthreads();
    }
    mx = sh[0];
    __syncthreads();
    float sum = 0.0f;
    for (int j = tid; j < L; j += blockDim.x) {
        float e = __expf(row[j] - mx);
        row[j] = e;
        sum += e;
    }
    sh[tid] = sum;
    __syncthreads();
    for (int off = 128; off > 0; off >>= 1) {
        if (tid < off) sh[tid] += sh[tid + off];
        __syncthreads();
    }
    float inv = 1.0f / sh[0];
    for (int j = tid; j < L; j += blockDim.x) row[j] *= inv;
}

// q = x[:, ::2, ::2, :]
__global__ void subsample_k(const float* __restrict__ X, float* __restrict__ Y,
                            int Bn, int H, int W, int C, int oh)
{
    long i = (long)blockIdx.x * blockDim.x + threadIdx.x;
    long total = (long)Bn * oh * oh * C;
    if (i >= total) return;
    int c = (int)(i % C);
    long t = i / C;
    int ox = (int)(t % oh); t /= oh;
    int oy = (int)(t % oh); t /= oh;
    int b  = (int)t;
    Y[i] = X[(((long)b * H + oy * 2) * W + ox * 2) * C + c];
}

__global__ void mean_pool_k(const float* __restrict__ X, float* __restrict__ Y,
                            int Bn, int N, int C)
{
    int i = blockIdx.x * blockDim.x + threadIdx.x;
    if (i >= Bn * C) return;
    int c = i % C, b = i / C;
    float s = 0.0f;
    for (int n = 0; n < N; n++) s += X[((long)b * N + n) * C + c];
    Y[i] = s / (float)N;
}

// ---------------------------------------------------------------------------
// Host-side orchestration
// ---------------------------------------------------------------------------
static inline unsigned cdiv(long a, long b) { return (unsigned)((a + b - 1) / b); }

static void gemmL(hipStream_t st, const float* A, const float* B, float* C,
                  int M, int N, int K, int lda, int ldb, int ldc,
                  int batch, int heads,
                  long sAb, long sAh, long sBb, long sBh, long sCb, long sCh,
                  int transB, float alpha,
                  const float* biasM, int biasLd, const float* biasV)
{
    dim3 grid(cdiv(N, TBN), cdiv(M, TBM), (unsigned)(batch * heads));
    gemm_f32_wmma<<<grid, 256, 0, st>>>(A, B, C, M, N, K, lda, ldb, ldc, heads,
                                        sAb, sAh, sBb, sBh, sCb, sCh,
                                        transB, alpha, biasM, biasLd, biasV);
}

static void elem_stats_bn(hipStream_t st, float* X, long M, int C,
                          float* mean, float* rstd,
                          const float* scale, const float* bias, int act)
{
    colstats<<<C, 256, 0, st>>>(X, (int)M, C, mean, rstd);
    long n = M * C;
    bn_act<<<cdiv(n, 256), 256, 0, st>>>(X, n, C, mean, rstd, scale, bias, act);
}

// mhdp attention block (residual handled here)
static void run_attn(hipStream_t st, float* x, int Bn, int N, int C, int heads, int dim,
                     const float* bias, const float* s_qkv,
                     const float* w_proj, const float* w_qkv,
                     float* qkv, float* scores, float* obuf, float* tbuf,
                     float* mean, float* rstd)
{
    int D = heads * dim, W4 = 4 * D, e = 2 * dim;
    long M = (long)Bn * N;
    float sc = 1.0f / sqrtf((float)dim);

    gemmL(st, x, w_qkv, qkv, (int)M, W4, C, C, W4, W4,
          1, 1, 0, 0, 0, 0, 0, 0, 0, 1.0f, nullptr, 0, nullptr);
    elem_stats_bn(st, qkv, M, W4, mean, rstd, s_qkv, nullptr, 0);

    // scores[b,h] = q @ k^T / sqrt(d) + bias
    gemmL(st, qkv, qkv + D, scores, N, N, dim, W4, W4, N,
          Bn, heads, (long)N * W4, dim, (long)N * W4, dim,
          (long)heads * N * N, (long)N * N,
          1, sc, bias, N, nullptr);
    softmax_rows<<<(unsigned)((long)Bn * heads * N), 256, 0, st>>>(scores, N);

    // o[b,h] = attn @ v  -> stored as [B, N, heads*e]
    gemmL(st, scores, qkv + 2 * D, obuf, N, e, N, N, W4, heads * e,
          Bn, heads, (long)heads * N * N, (long)N * N,
          (long)N * W4, e, (long)N * heads * e, e,
          0, 1.0f, nullptr, 0, nullptr);
    hswish_k<<<cdiv(M * heads * e, 256), 256, 0, st>>>(obuf, M * heads * e);

    gemmL(st, obuf, w_proj, tbuf, (int)M, C, heads * e, heads * e, C, C,
          1, 1, 0, 0, 0, 0, 0, 0, 0, 1.0f, nullptr, 0, nullptr);
    add_k<<<cdiv(M * C, 256), 256, 0, st>>>(x, tbuf, M * C);
}

static void run_ffn(hipStream_t st, float* x, long M, int C,
                    const float* b1, const float* b2,
                    const float* s1, const float* s2,
                    const float* w1, const float* w2,
                    float* hbuf, float* tbuf, float* mean, float* rstd)
{
    int Ch = 2 * C;
    gemmL(st, x, w1, hbuf, (int)M, Ch, C, C, Ch, Ch,
          1, 1, 0, 0, 0, 0, 0, 0, 0, 1.0f, nullptr, 0, nullptr);
    elem_stats_bn(st, hbuf, M, Ch, mean, rstd, s1, b1, 1);   // BN + hard-swish
    gemmL(st, hbuf, w2, tbuf, (int)M, C, Ch, Ch, C, C,
          1, 1, 0, 0, 0, 0, 0, 0, 0, 1.0f, nullptr, 0, nullptr);
    elem_stats_bn(st, tbuf, M, C, mean, rstd, s2, b2, 0);
    add_k<<<cdiv(M * C, 256), 256, 0, st>>>(x, tbuf, M * C);
}

static void run_sub(hipStream_t st, const float* x, float* xout,
                    int Bn, int H, int C, int heads, int dim, int outC,
                    const float* bias, const float* s_kv, const float* s_q,
                    const float* w_kv, const float* w_proj, const float* w_q,
                    float* kv, float* scores, float* qbuf, float* tbuf,
                    float* mean, float* rstd)
{
    int N = H * H, oh = (H - 1) / 2 + 1, No = oh * oh;
    int D = heads * dim, W5 = 5 * D, e = 4 * dim;
    long M = (long)Bn * N, Mo = (long)Bn * No;
    float sc = 1.0f / sqrtf((float)dim);

    gemmL(st, x, w_kv, kv, (int)M, W5, C, C, W5, W5,
          1, 1, 0, 0, 0, 0, 0, 0, 0, 1.0f, nullptr, 0, nullptr);
    elem_stats_bn(st, kv, M, W5, mean, rstd, s_kv, nullptr, 0);

    subsample_k<<<cdiv(Mo * C, 256), 256, 0, st>>>(x, tbuf, Bn, H, H, C, oh);
    gemmL(st, tbuf, w_q, qbuf, (int)Mo, D, C, C, D, D,
          1, 1, 0, 0, 0, 0, 0, 0, 0, 1.0f, nullptr, 0, nullptr);
    elem_stats_bn(st, qbuf, Mo, D, mean, rstd, s_q, nullptr, 0);

    gemmL(st, qbuf, kv, scores, No, N, dim, D, W5, N,
          Bn, heads, (long)No * D, dim, (long)N * W5, dim,
          (long)heads * No * N, (long)No * N,
          1, sc, bias, N, nullptr);
    softmax_rows<<<(unsigned)((long)Bn * heads * No), 256, 0, st>>>(scores, N);

    gemmL(st, scores, kv + D, tbuf, No, e, N, N, W5, heads * e,
          Bn, heads, (long)heads * No * N, (long)No * N,
          (long)N * W5, e, (long)No * heads * e, e,
          0, 1.0f, nullptr, 0, nullptr);
    hswish_k<<<cdiv(Mo * heads * e, 256), 256, 0, st>>>(tbuf, Mo * heads * e);

    gemmL(st, tbuf, w_proj, xout, (int)Mo, outC, heads * e, heads * e, outC, outC,
          1, 1, 0, 0, 0, 0, 0, 0, 0, 1.0f, nullptr, 0, nullptr);
}

extern "C" void kernel_launch(void* const* d_in, const int* in_sizes, int n_in,
                              void* d_out, int out_size, void* d_ws, size_t ws_size,
                              hipStream_t stream)
{
    (void)in_sizes; (void)n_in; (void)out_size; (void)ws_size;

    // Input order: x, then params flattened in JAX tree order (dict keys sorted
    // recursively, lists in index order):
    //   head_cls{b,w}, head_dis{b,w},
    //   stage1..3 blocks: attn{bias,s_qkv,w_proj,w_qkv}, ffn{b1,b2,s1,s2,w1,w2},
    //   stem[0..3]{b,w}, sub1{bias,s_kv,s_q,w_kv,w_proj,w_q}, sub2{same}
    int ci = 0;
    auto F = [&](int i) { return (const float*)d_in[i]; };
    const float* x_in = F(ci++);
    const float* hc_b = F(ci++); const float* hc_w = F(ci++);
    const float* hd_b = F(ci++); const float* hd_w = F(ci++);

    struct AttnP { const float *bias, *s_qkv, *w_proj, *w_qkv; };
    struct FfnP  { const float *b1, *b2, *s1, *s2, *w1, *w2; };
    AttnP at[9]; FfnP ff[9];
    const int depths[3] = {2, 3, 4};
    int nb = 0;
    for (int s = 0; s < 3; s++)
        for (int d = 0; d < depths[s]; d++) {
            at[nb].bias = F(ci++); at[nb].s_qkv = F(ci++);
            at[nb].w_proj = F(ci++); at[nb].w_qkv = F(ci++);
            ff[nb].b1 = F(ci++); ff[nb].b2 = F(ci++);
            ff[nb].s1 = F(ci++); ff[nb].s2 = F(ci++);
            ff[nb].w1 = F(ci++); ff[nb].w2 = F(ci++);
            nb++;
        }
    const float *st_b[4], *st_w[4];
    for (int i = 0; i < 4; i++) { st_b[i] = F(ci++); st_w[i] = F(ci++); }
    struct SubP { const float *bias, *s_kv, *s_q, *w_kv, *w_proj, *w_q; };
    SubP sub[2];
    for (int i = 0; i < 2; i++) {
        sub[i].bias = F(ci++); sub[i].s_kv = F(ci++); sub[i].s_q = F(ci++);
        sub[i].w_kv = F(ci++); sub[i].w_proj = F(ci++); sub[i].w_q = F(ci++);
    }

    // Workspace carve (floats)
    float* ws = (float*)d_ws;
    size_t off = 0;
    auto alloc = [&](size_t n) { float* p = ws + off; off += n; return p; };
    float* bufS  = alloc(13000000);  // stem ping-pong / attention scores (max 9.83M)
    float* bufA  = alloc(6600000);   // main activation x
    float* bufQ  = alloc(8600000);   // qkv / kv / ffn hidden (max 8.03M)
    float* bufO  = alloc(2100000);   // attention o / sub q
    float* bufT  = alloc(2100000);   // proj temp / subsampled q / sub o
    float* meanB = alloc(2048);
    float* rstdB = alloc(2048);
    float* pooled = alloc(64 * 384);

    const int Bn = 64;

    // ---- Stem: 4x (3x3, stride 2, SAME) convs: 3->16->32->64->128
    {
        long n1 = (long)Bn * 112 * 112 * 16;
        conv3x3s2<<<cdiv(n1, 256), 256, 0, stream>>>(x_in, st_w[0], st_b[0], bufS,
                                                     Bn, 224, 224, 3, 112, 112, 16);
        long n2 = (long)Bn * 56 * 56 * 32;
        conv3x3s2<<<cdiv(n2, 256), 256, 0, stream>>>(bufS, st_w[1], st_b[1], bufA,
                                                     Bn, 112, 112, 16, 56, 56, 32);
        long n3 = (long)Bn * 28 * 28 * 64;
        conv3x3s2<<<cdiv(n3, 256), 256, 0, stream>>>(bufA, st_w[2], st_b[2], bufS,
                                                     Bn, 56, 56, 32, 28, 28, 64);
        long n4 = (long)Bn * 14 * 14 * 128;
        conv3x3s2<<<cdiv(n4, 256), 256, 0, stream>>>(bufS, st_w[3], st_b[3], bufA,
                                                     Bn, 28, 28, 64, 14, 14, 128);
    }

    float* x = bufA;   // [64, 14, 14, 128]
    int blk = 0;

    // ---- Stage 1: C=128, N=196, heads=4
    for (int d = 0; d < 2; d++, blk++) {
        run_attn(stream, x, Bn, 196, 128, 4, 16,
                 at[blk].bias, at[blk].s_qkv, at[blk].w_proj, at[blk].w_qkv,
                 bufQ, bufS, bufO, bufT, meanB, rstdB);
        run_ffn(stream, x, (long)Bn * 196, 128,
                ff[blk].b1, ff[blk].b2, ff[blk].s1, ff[blk].s2, ff[blk].w1, ff[blk].w2,
                bufQ, bufT, meanB, rstdB);
    }
    // ---- Subsample 1: 14x14x128 -> 7x7x256, heads=8
    run_sub(stream, x, x, Bn, 14, 128, 8, 16, 256,
            sub[0].bias, sub[0].s_kv, sub[0].s_q, sub[0].w_kv, sub[0].w_proj, sub[0].w_q,
            bufQ, bufS, bufO, bufT, meanB, rstdB);

    // ---- Stage 2: C=256, N=49, heads=6
    for (int d = 0; d < 3; d++, blk++) {
        run_attn(stream, x, Bn, 49, 256, 6, 16,
                 at[blk].bias, at[blk].s_qkv, at[blk].w_proj, at[blk].w_qkv,
                 bufQ, bufS, bufO, bufT, meanB, rstdB);
        run_ffn(stream, x, (long)Bn * 49, 256,
                ff[blk].b1, ff[blk].b2, ff[blk].s1, ff[blk].s2, ff[blk].w1, ff[blk].w2,
                bufQ, bufT, meanB, rstdB);
    }
    // ---- Subsample 2: 7x7x256 -> 4x4x384, heads=16
    run_sub(stream, x, x, Bn, 7, 256, 16, 16, 384,
            sub[1].bias, sub[1].s_kv, sub[1].s_q, sub[1].w_kv, sub[1].w_proj, sub[1].w_q,
            bufQ, bufS, bufO, bufT, meanB, rstdB);

    // ---- Stage 3: C=384, N=16, heads=8
    for (int d = 0; d < 4; d++, blk++) {
        run_attn(stream, x, Bn, 16, 384, 8, 16,
                 at[blk].bias, at[blk].s_qkv, at[blk].w_proj, at[blk].w_qkv,
                 bufQ, bufS, bufO, bufT, meanB, rstdB);
        run_ffn(stream, x, (long)Bn * 16, 384,
                ff[blk].b1, ff[blk].b2, ff[blk].s1, ff[blk].s2, ff[blk].w1, ff[blk].w2,
                bufQ, bufT, meanB, rstdB);
    }

    // ---- Heads: mean pool + two linear classifiers (cls first, then dis)
    mean_pool_k<<<cdiv(Bn * 384, 256), 256, 0, stream>>>(x, pooled, Bn, 16, 384);
    float* out = (float*)d_out;
    gemmL(stream, pooled, hc_w, out, Bn, 1000, 384, 384, 1000, 1000,
          1, 1, 0, 0, 0, 0, 0, 0, 0, 1.0f, nullptr, 0, hc_b);
    gemmL(stream, pooled, hd_w, out + (long)Bn * 1000, Bn, 1000, 384, 384, 1000, 1000,
          1, 1, 0, 0, 0, 0, 0, 0, 0, 1.0f, nullptr, 0, hd_b);
}